// GINExpander_55027120996386
// MI455X (gfx1250) — compile-verified
//
#include <hip/hip_runtime.h>
#include <hip/hip_bf16.h>

#define N_NODES 50000
#define N_EDGES 800000
#define D 64
#define NUM_LAYERS 4
#define NUM_GRAPHS 128

typedef __attribute__((ext_vector_type(2))) float v2f;
typedef __attribute__((ext_vector_type(8))) float v8f;

// ---------------------------------------------------------------------------
// Elementwise copy (float4 vectorized): out = in.  n = number of float4s.
// ---------------------------------------------------------------------------
__global__ void copy_f4_kernel(const float4* __restrict__ in,
                               float4* __restrict__ out, int n) {
    int i = blockIdx.x * blockDim.x + threadIdx.x;
    if (i < n) out[i] = in[i];
}

// ---------------------------------------------------------------------------
// Zero a float buffer.
// ---------------------------------------------------------------------------
__global__ void zero_kernel(float* __restrict__ out, int n) {
    int i = blockIdx.x * blockDim.x + threadIdx.x;
    if (i < n) out[i] = 0.0f;
}

// ---------------------------------------------------------------------------
// Edge scatter-add: out[dst[e]][f] += x[src[e]][f].
// One thread per (edge, feature). Consecutive lanes = consecutive features of
// the same destination row -> atomics coalesce within a cacheline at L2.
// ---------------------------------------------------------------------------
__global__ void scatter_add_kernel(const float* __restrict__ x,
                                   const int* __restrict__ src,
                                   const int* __restrict__ dst,
                                   float* __restrict__ out) {
    int idx = blockIdx.x * blockDim.x + threadIdx.x;   // < 51.2M, fits int
    if (idx >= N_EDGES * D) return;
    int e = idx >> 6;
    int f = idx & (D - 1);
    int s = src[e];
    int d = dst[e];
    atomicAdd(out + (size_t)d * D + f, x[(size_t)s * D + f]);
}

// ---------------------------------------------------------------------------
// out = relu(h @ W^T + bias), h:[N_NODES,64] row-major, W:[64,64] row-major.
// Exact f32 tensor-core math via V_WMMA_F32_16X16X4_F32.
// Block = 128 threads (4 waves). Wave w computes the 16x16 tile
// rows [16*blockIdx.x, +16) x cols [16*w, +16), accumulating 16 K-steps of 4.
//
// A 16x4 layout : lanes 0-15 -> K={k0,k0+1}, lanes 16-31 -> K={k0+2,k0+3}
// B 4x16 layout : mirrored, lane = N column
// Both reduce to one aligned float2 load at (row*64 + k0 + 2*(lane>>4)).
// D layout      : col = lane&15, row = vgpr + 8*(lane>>4).
// EXEC is all 1s: no divergence, 50000 = 3125*16 exactly.
// ---------------------------------------------------------------------------
__global__ void gin_linear_relu_wmma(const float* __restrict__ h,
                                     const float* __restrict__ W,
                                     const float* __restrict__ bias,
                                     float* __restrict__ out) {
    const int lane = threadIdx.x & 31;
    const int wave = threadIdx.x >> 5;
    const int row0 = blockIdx.x * 16;
    const int col0 = wave * 16;
    const int m    = lane & 15;          // A: row within tile, B: col within tile
    const int koff = (lane >> 4) << 1;   // 0 for lanes 0-15, 2 for lanes 16-31

    const float* __restrict__ arow = h + (size_t)(row0 + m) * D;
    const float* __restrict__ brow = W + (size_t)(col0 + m) * D;

    v8f c = {};
#pragma unroll
    for (int k0 = 0; k0 < D; k0 += 4) {
        v2f a = *(const v2f*)(arow + k0 + koff);
        v2f b = *(const v2f*)(brow + k0 + koff);
        // D = A*B + C   (8 args: neg_a, A, neg_b, B, c_mod, C, reuse_a, reuse_b)
        c = __builtin_amdgcn_wmma_f32_16x16x4_f32(
                false, a, false, b, (short)0, c, false, false);
    }

    const float bc    = bias[col0 + m];
    const int   rbase = row0 + ((lane >> 4) << 3);
#pragma unroll
    for (int v = 0; v < 8; ++v) {
        float r = c[v] + bc;
        out[(size_t)(rbase + v) * D + col0 + m] = r > 0.0f ? r : 0.0f;
    }
}

// ---------------------------------------------------------------------------
// Graph pooling: out[batch[n]][f] += x[n][f].
// ---------------------------------------------------------------------------
__global__ void pool_kernel(const float* __restrict__ x,
                            const int* __restrict__ batch,
                            float* __restrict__ out) {
    int idx = blockIdx.x * blockDim.x + threadIdx.x;
    if (idx >= N_NODES * D) return;
    int n = idx >> 6;
    int f = idx & (D - 1);
    atomicAdd(out + (size_t)batch[n] * D + f, x[idx]);
}

// ---------------------------------------------------------------------------
extern "C" void kernel_launch(void* const* d_in, const int* in_sizes, int n_in,
                              void* d_out, int out_size, void* d_ws, size_t ws_size,
                              hipStream_t stream) {
    const float* x_in  = (const float*)d_in[0];
    const int*   ei    = (const int*)d_in[1];   // [2, N_EDGES] flat
    const int*   xei   = (const int*)d_in[2];   // [2, N_EDGES] flat
    const int*   batch = (const int*)d_in[3];   // [N_NODES]
    const float* Ws    = (const float*)d_in[4]; // [L, D, D]
    const float* bs    = (const float*)d_in[5]; // [L, D]
    float*       out   = (float*)d_out;         // [NUM_GRAPHS * D]

    const size_t buf_elems = (size_t)N_NODES * D;
    float* bufA = (float*)d_ws;                 // layer output x
    float* bufB = bufA + buf_elems;             // aggregated h
    float* bufC = bufB + buf_elems;             // relu(linear(h))

    const int n_f4       = N_NODES * D / 4;            // 800000
    const int copy_grid  = (n_f4 + 255) / 256;         // 3125
    const int scat_grid  = (N_EDGES * D + 255) / 256;  // 200000
    const int gemm_grid  = N_NODES / 16;               // 3125 (exact)
    const int pool_grid  = (N_NODES * D + 255) / 256;  // 12500

    const float* cur = x_in;
    for (int i = 0; i < NUM_LAYERS; ++i) {
        // h = (1+eps)*cur + scatter_sum(cur, real edges)   (eps = 0)
        copy_f4_kernel<<<copy_grid, 256, 0, stream>>>(
            (const float4*)cur, (float4*)bufB, n_f4);
        scatter_add_kernel<<<scat_grid, 256, 0, stream>>>(
            cur, ei, ei + N_EDGES, bufB);

        // h = relu(h @ W_i^T + b_i)    — f32 WMMA
        gin_linear_relu_wmma<<<gemm_grid, 128, 0, stream>>>(
            bufB, Ws + (size_t)i * D * D, bs + (size_t)i * D, bufC);

        // x = h + scatter_sum(h, expander edges)
        copy_f4_kernel<<<copy_grid, 256, 0, stream>>>(
            (const float4*)bufC, (float4*)bufA, n_f4);
        scatter_add_kernel<<<scat_grid, 256, 0, stream>>>(
            bufC, xei, xei + N_EDGES, bufA);

        cur = bufA;
    }

    // pooled[g] = sum over nodes in graph g
    zero_kernel<<<(NUM_GRAPHS * D + 255) / 256, 256, 0, stream>>>(
        out, NUM_GRAPHS * D);
    pool_kernel<<<pool_grid, 256, 0, stream>>>(cur, batch, out);
}